// PCEN_8572754723404
// MI455X (gfx1250) — compile-verified
//
#include <hip/hip_runtime.h>
#include <hip/hip_bf16.h>
#include <cstdint>

// PCEN: per-channel EMA over time + compressive nonlinearity.
// Dims fixed by the reference.
#define PB 128
#define PT 8000
#define PC 40
#define FLOOR_EPS 1e-6f

// ---- K1 chunking: fine-grained partial scans for parallelism ----
#define LC   100               // chunk length
#define NCH  (PT / LC)         // 80 chunks per chain

// ---- K3 blocking: one block per (batch, T-range) with LDS tile pipeline ----
#define NSPLIT 10
#define LRANGE (PT / NSPLIT)           // 800 timesteps per block
#define NSUB   16                      // sub-chunks per tile
#define SLEN   10                      // steps per sub-chunk per tile
#define TILE_T (NSUB * SLEN)           // 160 timesteps per tile
#define NTILES (LRANGE / TILE_T)       // 5 tiles per block
#define NTHR   (NSUB * PC)             // 640 threads per block
#define TILE_ELEMS (TILE_T * PC)       // 6400 floats = 25600 B per tile
#define CHUNKS_PER_SPLIT (LRANGE / LC) // 8

// Hardware transcendentals (v_log_f32 is log2, v_exp_f32 is exp2).
__device__ __forceinline__ float fast_log2(float x) { return __builtin_amdgcn_logf(x); }
__device__ __forceinline__ float fast_exp2(float x) { return __builtin_amdgcn_exp2f(x); }

// ------------------------------------------------------------------
// K1: each thread scans one LC-length chunk of one (b,c) chain with
// carry-in 0 and writes the local end state.  409,600 threads.
// ------------------------------------------------------------------
__global__ __launch_bounds__(256) void pcen_k1(const float* __restrict__ x,
                                               const float* __restrict__ smooth,
                                               float* __restrict__ E) {
  const int i = blockIdx.x * 256 + threadIdx.x;      // PB*NCH*PC total, exact
  const int c = i % PC;
  const int j = (i / PC) % NCH;
  const int b = i / (PC * NCH);
  const float w = fminf(fmaxf(smooth[c], 0.f), 1.f);
  const float d = 1.f - w;
  const float* xp = x + ((size_t)b * PT + (size_t)j * LC) * PC + c;
  float s = 0.f;
#pragma unroll 10
  for (int t = 0; t < LC; ++t)
    s = fmaf(d, s, w * xp[(size_t)t * PC]);
  E[i] = s;
}

// ------------------------------------------------------------------
// K2: per (b,c) chain, fold chunk partials (s_out = d^LC * s_in + E)
// and emit exact carry-ins at block (800-step) granularity.
// Carry for split 0 is x[b,0,c] (the reference scan initializer).
// ------------------------------------------------------------------
__global__ __launch_bounds__(256) void pcen_k2(const float* __restrict__ x,
                                               const float* __restrict__ smooth,
                                               const float* __restrict__ E,
                                               float* __restrict__ BC) {
  const int i = blockIdx.x * 256 + threadIdx.x;      // PB*PC total
  if (i >= PB * PC) return;
  const int c = i % PC;
  const int b = i / PC;
  const float w = fminf(fmaxf(smooth[c], 0.f), 1.f);
  const float d = 1.f - w;
  const float dL = powf(d, (float)LC);
  float car = x[(size_t)b * PT * PC + c];            // s_{-1} = x[b,0,c]
  const float* Eb = E + (size_t)b * NCH * PC + c;
  float* BCb = BC + (size_t)b * NSPLIT * PC + c;
#pragma unroll 8
  for (int j = 0; j < NCH; ++j) {
    if ((j % CHUNKS_PER_SPLIT) == 0)
      BCb[(size_t)(j / CHUNKS_PER_SPLIT) * PC] = car;
    car = fmaf(dL, car, Eb[(size_t)j * PC]);
  }
}

// ------------------------------------------------------------------
// Async global->LDS tile copy: tile is 25600 contiguous bytes = 3200
// 8-byte chunks; 640 threads issue 5 B64 async loads each.
// ------------------------------------------------------------------
__device__ __forceinline__ void async_load_tile(const float* __restrict__ gsrc,
                                                uint32_t lds_base, int tid) {
#pragma unroll
  for (int i = 0; i < (TILE_ELEMS * 4) / (8 * NTHR); ++i) {   // 5
    const uint32_t chunk = (uint32_t)(tid + i * NTHR);
    const uint32_t lds_addr = lds_base + chunk * 8u;
    const uint32_t goff = chunk * 8u;
    asm volatile("global_load_async_to_lds_b64 %0, %1, %2 offset:0"
                 :: "v"(lds_addr), "v"(goff), "s"(gsrc)
                 : "memory");
  }
}

// ------------------------------------------------------------------
// K3: one block per (b, split).  Double-buffered async LDS pipeline:
//   wait asynccnt -> barrier -> prefetch next tile -> pass A local
//   scans -> fold carries across 16 sub-chunks -> pass B rescan with
//   exact carry, PCEN math, non-temporal stores.
// ------------------------------------------------------------------
__global__ __launch_bounds__(NTHR) void pcen_k3(const float* __restrict__ x,
                                                const float* __restrict__ alpha_,
                                                const float* __restrict__ delta_,
                                                const float* __restrict__ root_,
                                                const float* __restrict__ smooth_,
                                                const float* __restrict__ BC,
                                                float* __restrict__ out) {
  __shared__ float xbuf[2][TILE_ELEMS];   // 2 x 25600 B
  __shared__ float Car[NSUB * PC];        // local ends -> carry-ins
  __shared__ float G[PC];                 // running carry per channel

  const int bs = blockIdx.x;              // b*NSPLIT + split
  const int b = bs / NSPLIT;
  const int split = bs % NSPLIT;
  const int tid = threadIdx.x;
  const int k = tid / PC;                 // sub-chunk 0..15
  const int c = tid % PC;                 // channel

  const float alpha = fminf(alpha_[c], 1.f);
  const float inv_root = 1.f / fmaxf(root_[c], 1.f);
  const float delta = delta_[c];
  const float w = fminf(fmaxf(smooth_[c], 0.f), 1.f);
  const float d = 1.f - w;
  const float dS = powf(d, (float)SLEN);
  const float delta_root = powf(delta, inv_root);

  const float* xr = x + ((size_t)b * PT + (size_t)split * LRANGE) * PC;
  float* orng = out + ((size_t)b * PT + (size_t)split * LRANGE) * PC;

  if (tid < PC) G[tid] = BC[(size_t)bs * PC + tid];

  const uint32_t lds0 = (uint32_t)(uintptr_t)(&xbuf[0][0]);
  const uint32_t lds1 = (uint32_t)(uintptr_t)(&xbuf[1][0]);

  async_load_tile(xr, lds0, tid);

  for (int tile = 0; tile < NTILES; ++tile) {
    const int cur = tile & 1;
    // Wait for this tile's async loads (each wave waits for its own),
    // then barrier so every wave's chunk has landed in LDS.
    asm volatile("s_wait_asynccnt 0x0" ::: "memory");
    __syncthreads();
    // Prefetch next tile into the other buffer; all pass-B reads of
    // that buffer completed before the barrier above.
    if (tile + 1 < NTILES)
      async_load_tile(xr + (size_t)(tile + 1) * TILE_ELEMS,
                      cur ? lds0 : lds1, tid);

    const float* xt = &xbuf[cur][k * SLEN * PC + c];

    // Pass A: local scan with carry-in 0.
    float s = 0.f;
#pragma unroll
    for (int t = 0; t < SLEN; ++t)
      s = fmaf(d, s, w * xt[t * PC]);
    Car[k * PC + c] = s;
    __syncthreads();

    // Fold: one thread per channel chains 16 sub-chunk partials,
    // overwriting each slot with its exact carry-in.
    if (k == 0) {
      float car = G[c];
#pragma unroll
      for (int j = 0; j < NSUB; ++j) {
        const float e = Car[j * PC + c];
        Car[j * PC + c] = car;
        car = fmaf(dS, car, e);
      }
      G[c] = car;
    }
    __syncthreads();

    // Pass B: rescan with exact carry, apply PCEN, NT store.
    s = Car[k * PC + c];
    float* op = orng + (size_t)(tile * TILE_T + k * SLEN) * PC + c;
#pragma unroll
    for (int t = 0; t < SLEN; ++t) {
      const float xv = xt[t * PC];
      s = fmaf(d, s, w * xv);                               // ema
      const float p1 = fast_exp2(alpha * fast_log2(FLOOR_EPS + s));
      const float o = fast_exp2(inv_root * fast_log2(xv / p1 + delta)) - delta_root;
      __builtin_nontemporal_store(o, op + (size_t)t * PC);
    }
  }
}

extern "C" void kernel_launch(void* const* d_in, const int* in_sizes, int n_in,
                              void* d_out, int out_size, void* d_ws, size_t ws_size,
                              hipStream_t stream) {
  (void)in_sizes; (void)n_in; (void)out_size; (void)ws_size;
  const float* x      = (const float*)d_in[0];
  const float* alpha  = (const float*)d_in[1];
  const float* delta  = (const float*)d_in[2];
  const float* root   = (const float*)d_in[3];
  const float* smooth = (const float*)d_in[4];
  float* out = (float*)d_out;

  float* E  = (float*)d_ws;                        // PB*NCH*PC floats (1.6 MB)
  float* BC = E + (size_t)PB * NCH * PC;           // PB*NSPLIT*PC floats (205 KB)

  pcen_k1<<<(PB * NCH * PC) / 256, 256, 0, stream>>>(x, smooth, E);
  pcen_k2<<<(PB * PC + 255) / 256, 256, 0, stream>>>(x, smooth, E, BC);
  pcen_k3<<<PB * NSPLIT, NTHR, 0, stream>>>(x, alpha, delta, root, smooth, BC, out);
}